// FocalLoss_9569187136155
// MI455X (gfx1250) — compile-verified
//
#include <hip/hip_runtime.h>
#include <cstdint>

#define A_N 100000
#define B_N 8
#define C_N 80
#define M_N 32
#define TILE_A 64
#define BLOCK 256
#define ROW_PAD 84   // floats per padded LDS row (80 data + 4 pad for bank spread)

typedef float v2f __attribute__((ext_vector_type(2)));
typedef float v8f __attribute__((ext_vector_type(8)));
typedef int   v4i __attribute__((ext_vector_type(4)));

typedef __attribute__((address_space(1))) v4i* global_v4i_ptr;
typedef __attribute__((address_space(3))) v4i* lds_v4i_ptr;

#define LN2F 0.6931471805599453f

// fast natural log: v_log_f32 (log2) * ln2 — inputs are clamped away from 0/Inf
__device__ __forceinline__ float fast_log(float x) {
#if __has_builtin(__builtin_amdgcn_logf)
  return __builtin_amdgcn_logf(x) * LN2F;
#else
  return __logf(x);
#endif
}

// ---- CDNA5 async global->LDS DMA (ASYNCcnt-tracked) ----------------------
__device__ __forceinline__ void async_load_b128(const float* g, float* lds) {
#if __has_builtin(__builtin_amdgcn_global_load_async_to_lds_b128)
  float* gm = const_cast<float*>(g);
  __builtin_amdgcn_global_load_async_to_lds_b128(
      (global_v4i_ptr)gm, (lds_v4i_ptr)lds, 0, 0);
#else
  uint32_t l = (uint32_t)(uintptr_t)lds;   // low 32 bits of flat shared addr == LDS offset
  asm volatile("global_load_async_to_lds_b128 %0, %1, off"
               :: "v"(l), "v"(g) : "memory");
#endif
}

__device__ __forceinline__ void wait_async0() {
#if __has_builtin(__builtin_amdgcn_s_wait_asynccnt)
  __builtin_amdgcn_s_wait_asynccnt(0);
#else
  asm volatile("s_wait_asynccnt 0" ::: "memory");
#endif
}

// ---- wave32 f32 sum entirely on the matrix pipe --------------------------
// WMMA #1, A = {s,0}, B = ones(4x16): D[m][n] = s_m + s_{m+16}. Summing D's
// 8 VGPRs leaves R0 (sum of rows 0-7) replicated in lanes 0-15 and R1 (rows
// 8-15) in lanes 16-31. WMMA #2 on that row-replicated value yields
// D2[m][n] = r_m + r_{m+16} = R0 + R1 = full wave sum, in every lane —
// no ds_bpermute / cross-lane shuffle needed, exact in fp32.
__device__ __forceinline__ float wave_reduce_sum(float s) {
#if __has_builtin(__builtin_amdgcn_wmma_f32_16x16x4_f32)
  v2f ones; ones[0] = 1.0f; ones[1] = 1.0f;
  v2f a;    a[0] = s;    a[1] = 0.0f;
  v8f acc = {0.f,0.f,0.f,0.f,0.f,0.f,0.f,0.f};
  acc = __builtin_amdgcn_wmma_f32_16x16x4_f32(false, a, false, ones,
                                              (short)0, acc, false, false);
  float r = ((acc[0]+acc[1])+(acc[2]+acc[3]))+((acc[4]+acc[5])+(acc[6]+acc[7]));
  v2f a2;   a2[0] = r;   a2[1] = 0.0f;
  v8f acc2 = {0.f,0.f,0.f,0.f,0.f,0.f,0.f,0.f};
  acc2 = __builtin_amdgcn_wmma_f32_16x16x4_f32(false, a2, false, ones,
                                               (short)0, acc2, false, false);
  return acc2[0];
#else
  for (int off = 16; off >= 1; off >>= 1) s += __shfl_xor(s, off, 32);
  return s;
#endif
}

__global__ void focal_init(float* gCls, float* gReg, unsigned* gCnt) {
  int t = threadIdx.x;
  if (t < B_N) { gCls[t] = 0.0f; gReg[t] = 0.0f; gCnt[t] = 0u; }
}

__global__ __launch_bounds__(BLOCK)
void focal_main(const float* __restrict__ cls, const float* __restrict__ reg,
                const float* __restrict__ anc, const float* __restrict__ ann,
                float* __restrict__ gCls, float* __restrict__ gReg,
                unsigned* __restrict__ gCnt)
{
  __shared__ __align__(16) float clsTile[TILE_A * ROW_PAD];
  __shared__ float   boxes[M_N * 5];
  __shared__ int     state[TILE_A];
  __shared__ float   waveSum[BLOCK / 32];
  __shared__ float   sReg;
  __shared__ unsigned sCnt;

  const int b   = blockIdx.y;
  const int a0  = blockIdx.x * TILE_A;
  const int tid = threadIdx.x;

  // (1) launch async DMA of the 64x80 class tile into LDS (row-padded)
  const float* clsImg = cls + (size_t)b * A_N * C_N;
  #pragma unroll
  for (int i = 0; i < 5; ++i) {
    int idx = tid + BLOCK * i;          // 0..1279 = 64 rows * 20 chunks
    int r = idx / 20, c = idx % 20;
    int aRow = a0 + r; if (aRow >= A_N) aRow = A_N - 1;   // clamp (masked later)
    async_load_b128(clsImg + (size_t)aRow * C_N + c * 4,
                    &clsTile[r * ROW_PAD + c * 4]);
  }

  // (2) boxes + accumulator init
  if (tid < M_N * 5) boxes[tid] = ann[(size_t)b * M_N * 5 + tid];
  if (tid == 0) { sReg = 0.0f; sCnt = 0u; }
  __syncthreads();

  // (3) Phase A: IoU assignment per anchor (overlaps with DMA in flight)
  if (tid < TILE_A) {
    int a  = a0 + tid;
    int st = -1;                                   // ignore / out-of-range
    if (a < A_N) {
      float ay1 = anc[a*4+0], ax1 = anc[a*4+1], ay2 = anc[a*4+2], ax2 = anc[a*4+3];
      float aw = ax2 - ax1, ah = ay2 - ay1;
      float aArea = aw * ah;
      float best = -1.0f; int bm = 0;
      for (int m = 0; m < M_N; ++m) {
        float bx1 = boxes[m*5+0], by1 = boxes[m*5+1];
        float bx2 = boxes[m*5+2], by2 = boxes[m*5+3];
        float iw = fminf(ax2, bx2) - fmaxf(ax1, bx1);
        float ih = fminf(ay2, by2) - fmaxf(ay1, by1);
        iw = fmaxf(iw, 0.0f); ih = fmaxf(ih, 0.0f);
        float inter = iw * ih;
        float ua  = fmaxf(aArea + (bx2 - bx1) * (by2 - by1) - inter, 1e-8f);
        float iou = inter / ua;
        if (iou > best) { best = iou; bm = m; }    // first-max like jnp.argmax
      }
      if (best >= 0.5f) {                          // positive anchor
        st = 1 + (int)boxes[bm*5+4];
        float bx1 = boxes[bm*5+0], by1 = boxes[bm*5+1];
        float bx2 = boxes[bm*5+2], by2 = boxes[bm*5+3];
        float gwr = bx2 - bx1, ghr = by2 - by1;
        float gcx = bx1 + 0.5f * gwr, gcy = by1 + 0.5f * ghr;
        float gw  = fmaxf(gwr, 1.0f), gh = fmaxf(ghr, 1.0f);
        float acx = ax1 + 0.5f * aw, acy = ay1 + 0.5f * ah;
        float tv[4] = { (gcy - acy) / ah, (gcx - acx) / aw,
                        fast_log(gh / ah), fast_log(gw / aw) };
        const float* rg = reg + ((size_t)b * A_N + a) * 4;
        float rl = 0.0f;
        #pragma unroll
        for (int k = 0; k < 4; ++k) {
          float d = fabsf(tv[k] - rg[k]);
          rl += (d <= (1.0f / 9.0f)) ? 4.5f * d * d : (d - 1.0f / 18.0f);
        }
        atomicAdd(&sReg, rl);
        atomicAdd(&sCnt, 1u);
      } else if (best < 0.4f) {
        st = 0;                                    // negative: all-zero targets
      }
    }
    state[tid] = st;
  }

  // (4) DMA completion + visibility of state/boxes across waves
  wait_async0();
  __syncthreads();

  // (5) Phase B: focal loss, 4 threads/anchor, 20 classes each, from LDS
  int r = tid >> 2, sub = tid & 3;
  int st = state[r];
  float s = 0.0f;
  if (st >= 0) {
    int posCls = st - 1;                           // -1 when negative anchor
    #pragma unroll
    for (int i = 0; i < 5; ++i) {
      int q = sub + 4 * i;                         // chunk 0..19
      const float4 v = *reinterpret_cast<const float4*>(&clsTile[r * ROW_PAD + q * 4]);
      float pv[4] = { v.x, v.y, v.z, v.w };
      #pragma unroll
      for (int j = 0; j < 4; ++j) {
        int  cIdx = q * 4 + j;
        float p   = fminf(fmaxf(pv[j], 1e-4f), 1.0f - 1e-4f);
        bool  is1 = (cIdx == posCls);
        float pt  = is1 ? (1.0f - p) : p;
        float w   = (is1 ? 0.25f : 0.75f) * pt * pt;      // alpha * (.)^gamma
        float bce = -fast_log(is1 ? p : (1.0f - p));
        s += w * bce;
      }
    }
  }

  // (6) wave reduce on the matrix pipe, block combine, one atomic per image
  float ws = wave_reduce_sum(s);                   // EXEC all ones here
  int wave = tid >> 5;
  if ((tid & 31) == 0) waveSum[wave] = ws;
  __syncthreads();
  if (tid == 0) {
    float tot = 0.0f;
    #pragma unroll
    for (int wv = 0; wv < BLOCK / 32; ++wv) tot += waveSum[wv];
    atomicAdd(&gCls[b], tot);
    atomicAdd(&gReg[b], sReg);
    atomicAdd(&gCnt[b], sCnt);
  }
}

__global__ void focal_final(const float* gCls, const float* gReg,
                            const unsigned* gCnt, float* out) {
  if (threadIdx.x == 0) {
    float cm = 0.0f, rm = 0.0f;
    for (int bb = 0; bb < B_N; ++bb) {
      unsigned n = gCnt[bb];
      float np = fmaxf((float)n, 1.0f);
      cm += gCls[bb] / np;
      rm += (n > 0u) ? (gReg[bb] / (float)(4u * n)) : 0.0f;
    }
    cm *= (1.0f / B_N);
    rm *= (1.0f / B_N);
    out[0] = cm + rm;
    out[1] = cm;
    out[2] = rm;
  }
}

extern "C" void kernel_launch(void* const* d_in, const int* in_sizes, int n_in,
                              void* d_out, int out_size, void* d_ws, size_t ws_size,
                              hipStream_t stream) {
  const float* cls = (const float*)d_in[0];   // [B,A,C]
  const float* reg = (const float*)d_in[1];   // [B,A,4]
  const float* anc = (const float*)d_in[2];   // [1,A,4]
  const float* ann = (const float*)d_in[3];   // [B,M,5]
  float*    gCls = (float*)d_ws;
  float*    gReg = gCls + B_N;
  unsigned* gCnt = (unsigned*)(gReg + B_N);

  focal_init<<<1, 32, 0, stream>>>(gCls, gReg, gCnt);
  dim3 grid((A_N + TILE_A - 1) / TILE_A, B_N);
  focal_main<<<grid, BLOCK, 0, stream>>>(cls, reg, anc, ann, gCls, gReg, gCnt);
  focal_final<<<1, 32, 0, stream>>>(gCls, gReg, gCnt, (float*)d_out);
}